// SMPLLossPixel_64072322121837
// MI455X (gfx1250) — compile-verified
//
#include <hip/hip_runtime.h>

#define SCALE 64
#define S2 4096
#define THREADS 256
#define PIX_PER_BLOCK 128
#define BLOCKS_PER_BATCH (S2 / PIX_PER_BLOCK) /* 32 */

typedef __attribute__((ext_vector_type(2))) float v2f;
typedef __attribute__((ext_vector_type(4))) float f4v;
typedef __attribute__((ext_vector_type(8))) float v8f;
typedef __attribute__((ext_vector_type(4))) int   v4i;

#if defined(__gfx1250__) && __has_builtin(__builtin_amdgcn_wmma_f32_16x16x4_f32)
#define USE_WMMA 1
#else
#define USE_WMMA 0
#endif

#if defined(__gfx1250__) && __has_builtin(__builtin_amdgcn_global_load_async_to_lds_b128)
#define USE_ASYNC 1
#else
#define USE_ASYNC 0
#endif

__device__ __forceinline__ int clampi(int v, int lo, int hi) {
  return v < lo ? lo : (v > hi ? hi : v);
}

// Extract d[di] with a constant-index select tree (avoids scratch waterfall).
__device__ __forceinline__ float diag_extract(v8f d, int di) {
  float lo = (di & 2) ? ((di & 1) ? d[3] : d[2]) : ((di & 1) ? d[1] : d[0]);
  float hi = (di & 2) ? ((di & 1) ? d[7] : d[6]) : ((di & 1) ? d[5] : d[4]);
  return (di & 4) ? hi : lo;
}

__global__ __launch_bounds__(THREADS) void smpl_warp_wmma(
    const float* __restrict__ corr,       // (B, 4096, 4096)
    const float* __restrict__ gt_flow,    // (B, 2, 64, 64)
    const float* __restrict__ scale_ref,  // (B, 3, 64, 64)
    float* __restrict__ out_smpl,         // (B, 3, 64, 64)
    float* __restrict__ out_corr)         // (B, 3, 64, 64)
{
  __shared__ float ldsRef[3 * S2];  // 48 KB: scale_ref for this batch

  const int tid  = threadIdx.x;
  const int b    = blockIdx.x / BLOCKS_PER_BATCH;
  const int tile = blockIdx.x % BLOCKS_PER_BATCH;

  const float* gRef = scale_ref + (size_t)b * 3 * S2;

  // ---- Stage scale_ref[b] into LDS (CDNA5 async global->LDS path) ----
#if USE_ASYNC
  {
    const f4v* gsrc = (const f4v*)gRef;
    f4v*       ldst = (f4v*)ldsRef;
    for (int i = tid; i < (3 * S2) / 4; i += THREADS) {  // 12 rounds, uniform
      __builtin_amdgcn_global_load_async_to_lds_b128(
          (__attribute__((address_space(1))) v4i*)(gsrc + i),
          (__attribute__((address_space(3))) v4i*)(ldst + i),
          0, 0);
    }
#if __has_builtin(__builtin_amdgcn_s_wait_asynccnt)
    __builtin_amdgcn_s_wait_asynccnt(0);
#else
    asm volatile("s_wait_asynccnt 0x0" ::: "memory");
#endif
  }
#else
  for (int i = tid; i < 3 * S2; i += THREADS) ldsRef[i] = gRef[i];
#endif
  __syncthreads();

  // ---- Per-lane bilinear setup ----
  // Wave handles 16 pixels; lane L: pixel q = L&15, half h = L>>4.
  // h=0 covers neighbors k=0,1 (row fy), h=1 covers k=2,3 (row fy+1).
  const int lane = tid & 31;
  const int wave = tid >> 5;
  const int q    = lane & 15;
  const int h    = lane >> 4;
  const int pixg = tile * PIX_PER_BLOCK + wave * 16 + q;  // pixel within batch

  const float x  = gt_flow[(size_t)(b * 2 + 0) * S2 + pixg];
  const float y  = gt_flow[(size_t)(b * 2 + 1) * S2 + pixg];
  const float gx = (x + 1.0f) * ((SCALE - 1) * 0.5f);
  const float gy = (y + 1.0f) * ((SCALE - 1) * 0.5f);
  const float fx = floorf(gx);
  const float fy = floorf(gy);

  const int ifx  = (int)fx;
  const int ify  = (int)fy;
  const int row  = clampi(ify + h, 0, SCALE - 1);
  const int colA = clampi(ifx,     0, SCALE - 1);
  const int colB = clampi(ifx + 1, 0, SCALE - 1);
  const int idxA = row * SCALE + colA;
  const int idxB = row * SCALE + colB;

  const float dy = h ? (gy - fy) : (fy + 1.0f - gy);
  const float wA = dy * (fx + 1.0f - gx);  // k = 2h
  const float wB = dy * (gx - fx);         // k = 2h+1

  // Gather the only 2 corr elements this lane needs (row p, cols idxA/idxB).
  const float* crow = corr + ((size_t)b * S2 + (size_t)pixg) * S2;
  const float cA = crow[idxA];
  const float cB = crow[idxB];

#if USE_WMMA
  // A (16x4 f32): VGPR0 = K0|K2, VGPR1 = K1|K3 across lane halves -> per-lane
  // {wA,wB} / {cA,cB} is already the correct layout. B (4x16) mirrors it.
  const v2f a_w = {wA, wB};
  const v2f a_c = {cA, cB};

  const bool writer = (lane < 8) || (lane >= 24);
  const int  di     = (lane < 8) ? lane : (lane - 24);  // accumulator VGPR idx
  const int  pout   = (lane < 8) ? lane : (lane - 16);  // diagonal pixel
  const size_t opix = (size_t)tile * PIX_PER_BLOCK + (size_t)wave * 16 + pout;

#pragma unroll
  for (int c = 0; c < 3; ++c) {
    const v2f bm = {ldsRef[c * S2 + idxA], ldsRef[c * S2 + idxB]};
    v8f z = {};
    // D = A(16x4) * B(4x16); diagonal D[q,q] = exact per-pixel 4-dot.
    v8f dS = __builtin_amdgcn_wmma_f32_16x16x4_f32(
        false, a_w, false, bm, (short)0, z, false, false);
    v8f dC = __builtin_amdgcn_wmma_f32_16x16x4_f32(
        false, a_c, false, bm, (short)0, z, false, false);
    if (writer) {
      const size_t o = ((size_t)b * 3 + c) * S2 + opix;
      out_smpl[o] = diag_extract(dS, di);
      out_corr[o] = diag_extract(dC, di);
    }
  }
#else
  // VALU fallback: partial dot per half-wave, combine across halves.
#pragma unroll
  for (int c = 0; c < 3; ++c) {
    const float rA = ldsRef[c * S2 + idxA];
    const float rB = ldsRef[c * S2 + idxB];
    float ps = wA * rA + wB * rB;
    float pc = cA * rA + cB * rB;
    ps += __shfl_xor(ps, 16, 32);
    pc += __shfl_xor(pc, 16, 32);
    if (h == 0) {
      const size_t o = ((size_t)b * 3 + c) * S2 + (size_t)pixg;
      out_smpl[o] = ps;
      out_corr[o] = pc;
    }
  }
#endif
}

extern "C" void kernel_launch(void* const* d_in, const int* in_sizes, int n_in,
                              void* d_out, int out_size, void* d_ws, size_t ws_size,
                              hipStream_t stream) {
  const float* corr      = (const float*)d_in[0];  // (B, S2, S2)
  const float* gt_flow   = (const float*)d_in[1];  // (B, 2, 64, 64)
  // d_in[2] = vis_mask : unused by the reference math
  const float* scale_ref = (const float*)d_in[3];  // (B, 3, 64, 64)

  const int B = in_sizes[1] / (2 * S2);
  float* out_smpl = (float*)d_out;
  float* out_corr = out_smpl + (size_t)B * 3 * S2;

  dim3 grid(B * BLOCKS_PER_BATCH);
  dim3 block(THREADS);
  smpl_warp_wmma<<<grid, block, 0, stream>>>(corr, gt_flow, scale_ref,
                                             out_smpl, out_corr);
}